// DglRelationGraphConvolutionNetwork_21715354648951
// MI455X (gfx1250) — compile-verified
//
#include <hip/hip_runtime.h>

// R-GCN basis-decomposition forward for MI455X (gfx1250, wave32, WMMA).
//   t[n,b,o]  = sum_i h[n,i] * bases[b,i,o]          (WMMA f32 GEMM, 12.9 GFLOP)
//   msg[e,:]  = sum_b comp[rel[e],b] * t[src[e],b,:] (L2-resident gather)
//   agg[d,:] += msg[e,:]                             (global_atomic_add_f32)
//   out       = relu(agg + bias)
// d_ws holds t: N*3*256 floats = 100,663,296 bytes.
//
// GEMM kernel: 256 threads = 8 waves. Block owns (basis b, 64-column group).
// B panel is staged into LDS in exact WMMA-fragment order via async
// global->LDS loads; 8 waves each compute a 16x64 slice (4 accumulators).

typedef float v2f  __attribute__((ext_vector_type(2)));
typedef float v8f  __attribute__((ext_vector_type(8)));

#define IN_F  256
#define OUT_F 256
#define NBASE 3
#define KSTEPS (IN_F / 4)   // 64 wmma k-steps

#if __has_builtin(__builtin_amdgcn_global_load_async_to_lds_b32)
#define HAVE_ASYNC_LDS 1
typedef __attribute__((address_space(1))) int gbl_i32;   // clang prints AS1 as "__device__"
typedef __attribute__((address_space(3))) int lds_i32;
static __device__ __forceinline__ void async_copy_b32(const float* g, void* l) {
    __builtin_amdgcn_global_load_async_to_lds_b32((gbl_i32*)g, (lds_i32*)l, 0, 0);
}
#else
#define HAVE_ASYNC_LDS 0
#endif

static __device__ __forceinline__ void wait_asynccnt0() {
#if __has_builtin(__builtin_amdgcn_s_wait_asynccnt)
    __builtin_amdgcn_s_wait_asynccnt(0);
#else
    asm volatile("s_wait_asynccnt 0" ::: "memory");
#endif
}

// Grid: (N/128) row-groups x 4 col-groups x 3 bases. 8 waves = 8 row tiles.
__global__ void __launch_bounds__(256) rgcn_basis_gemm(
    const float* __restrict__ h,      // [N, 256]
    const float* __restrict__ bases,  // [3, 256, 256] (IN x OUT row-major)
    float* __restrict__ t)            // [N, 3, 256]
{
    // B fragments in WMMA lane order: Bfrag[kstep][colTile][lane] = {B[kA][col], B[kA+1][col]}
    // where kA = 4*kstep + 2*(lane>=16), col = ct*16 + (lane&15).  64 KB LDS.
    __shared__ float2 Bfrag[KSTEPS][4][32];

    const int tid     = threadIdx.x;
    const int lane    = tid & 31;
    const int ln      = lane & 15;
    const int halfSel = lane >> 4;

    int bid = blockIdx.x;
    const int bb = bid % NBASE;  bid /= NBASE;
    const int cg = bid & 3;                 // 64-column group
    const int rg = bid >> 2;                // row group (8 tiles of 16 rows)

    // ---- Stage B panel (256 x 64 f32) into LDS in fragment order ----
    const float* Bsrc = bases + (size_t)bb * (IN_F * OUT_F) + cg * 64;
    #pragma unroll 4
    for (int it = 0; it < 32; ++it) {
        const int combo = (tid >> 5) + it * 8;      // 0..255 -> (kstep, ct)
        const int s  = combo >> 2;
        const int ct = combo & 3;
        const int kA = 4 * s + 2 * halfSel;
        const float* g0 = Bsrc + (size_t)kA * OUT_F + ct * 16 + ln;
        float2* lp = &Bfrag[s][ct][lane];
#if HAVE_ASYNC_LDS
        async_copy_b32(g0,         (char*)lp + 0);
        async_copy_b32(g0 + OUT_F, (char*)lp + 4);
#else
        *lp = make_float2(g0[0], g0[OUT_F]);
#endif
    }
#if HAVE_ASYNC_LDS
    wait_asynccnt0();
#endif
    __syncthreads();

    // ---- Compute: this wave's 16x64 output slice ----
    const int rowTile = rg * 8 + (tid >> 5);
    const float* Arow = h + (size_t)(rowTile * 16 + ln) * IN_F + 2 * halfSel;

    v8f c0 = {}, c1 = {}, c2 = {}, c3 = {};
    #pragma unroll 4
    for (int s = 0; s < KSTEPS; ++s) {
        const v2f a  = *(const v2f*)(Arow + 4 * s);        // global b64 (L1/L2 hit)
        const v2f b0 = *(const v2f*)&Bfrag[s][0][lane];    // ds b64, conflict-free
        const v2f b1 = *(const v2f*)&Bfrag[s][1][lane];
        const v2f b2 = *(const v2f*)&Bfrag[s][2][lane];
        const v2f b3 = *(const v2f*)&Bfrag[s][3][lane];
        c0 = __builtin_amdgcn_wmma_f32_16x16x4_f32(false, a, false, b0, (short)0, c0, false, false);
        c1 = __builtin_amdgcn_wmma_f32_16x16x4_f32(false, a, false, b1, (short)0, c1, false, false);
        c2 = __builtin_amdgcn_wmma_f32_16x16x4_f32(false, a, false, b2, (short)0, c2, false, false);
        c3 = __builtin_amdgcn_wmma_f32_16x16x4_f32(false, a, false, b3, (short)0, c3, false, false);
    }

    // D layout: VGPR r -> row (r + 8*halfSel), col = ln (+16 per col tile).
    const int colBase = cg * 64 + ln;
    #pragma unroll
    for (int r = 0; r < 8; ++r) {
        const int m = rowTile * 16 + r + 8 * halfSel;
        float* row = t + (size_t)m * (NBASE * OUT_F) + bb * OUT_F + colBase;
        row[0]  = c0[r];
        row[16] = c1[r];
        row[32] = c2[r];
        row[48] = c3[r];
    }
}

// One wave per edge: gather t[src] (L2-resident), mix with comp[rel], atomically
// scatter 256 floats into agg[dst]. Each lane owns 8 contiguous output floats.
__global__ void __launch_bounds__(256) rgcn_edge_scatter(
    const float* __restrict__ t,     // [N, 3, 256]
    const int*   __restrict__ src,
    const int*   __restrict__ dst,
    const int*   __restrict__ rel,
    const float* __restrict__ comp,  // [R, 3]
    float*       __restrict__ agg,   // [N, 256] (pre-zeroed)
    int E)
{
    const int e = blockIdx.x * 8 + (threadIdx.x >> 5);
    if (e >= E) return;              // uniform per wave
    const int lane = threadIdx.x & 31;

    const int s = src[e], d = dst[e], r = rel[e];
    const float c0 = comp[r * NBASE + 0];
    const float c1 = comp[r * NBASE + 1];
    const float c2 = comp[r * NBASE + 2];

    const float* ts = t + (size_t)s * (NBASE * OUT_F);
    const int o = lane * 8;

    float4 t00 = *(const float4*)(ts + 0 * OUT_F + o);
    float4 t01 = *(const float4*)(ts + 0 * OUT_F + o + 4);
    float4 t10 = *(const float4*)(ts + 1 * OUT_F + o);
    float4 t11 = *(const float4*)(ts + 1 * OUT_F + o + 4);
    float4 t20 = *(const float4*)(ts + 2 * OUT_F + o);
    float4 t21 = *(const float4*)(ts + 2 * OUT_F + o + 4);

    float acc[8];
    acc[0] = c0 * t00.x + c1 * t10.x + c2 * t20.x;
    acc[1] = c0 * t00.y + c1 * t10.y + c2 * t20.y;
    acc[2] = c0 * t00.z + c1 * t10.z + c2 * t20.z;
    acc[3] = c0 * t00.w + c1 * t10.w + c2 * t20.w;
    acc[4] = c0 * t01.x + c1 * t11.x + c2 * t21.x;
    acc[5] = c0 * t01.y + c1 * t11.y + c2 * t21.y;
    acc[6] = c0 * t01.z + c1 * t11.z + c2 * t21.z;
    acc[7] = c0 * t01.w + c1 * t11.w + c2 * t21.w;

    float* ad = agg + (size_t)d * OUT_F + o;
    #pragma unroll
    for (int j = 0; j < 8; ++j) atomicAdd(ad + j, acc[j]);
}

__global__ void __launch_bounds__(256) rgcn_zero(float4* __restrict__ p, int n4)
{
    const int i = blockIdx.x * 256 + threadIdx.x;
    if (i < n4) p[i] = make_float4(0.f, 0.f, 0.f, 0.f);
}

__global__ void __launch_bounds__(256) rgcn_bias_relu(
    float* __restrict__ agg, const float* __restrict__ bias, int n4)
{
    const int i = blockIdx.x * 256 + threadIdx.x;
    if (i >= n4) return;
    float4 v = ((float4*)agg)[i];
    const int ob = (i * 4) & (OUT_F - 1);
    v.x = fmaxf(v.x + bias[ob + 0], 0.f);
    v.y = fmaxf(v.y + bias[ob + 1], 0.f);
    v.z = fmaxf(v.z + bias[ob + 2], 0.f);
    v.w = fmaxf(v.w + bias[ob + 3], 0.f);
    ((float4*)agg)[i] = v;
}

extern "C" void kernel_launch(void* const* d_in, const int* in_sizes, int n_in,
                              void* d_out, int out_size, void* d_ws, size_t ws_size,
                              hipStream_t stream)
{
    const float* text  = (const float*)d_in[0];
    const int*   src   = (const int*)  d_in[1];
    const int*   dst   = (const int*)  d_in[2];
    const int*   rel   = (const int*)  d_in[3];
    const float* bases = (const float*)d_in[4];
    const float* comp  = (const float*)d_in[5];
    const float* bias  = (const float*)d_in[6];
    float* out = (float*)d_out;
    float* t   = (float*)d_ws;                 // N*3*256 floats

    const int E = in_sizes[1];
    const int N = in_sizes[0] / IN_F;          // 32768; multiple of 128 by construction
    const int n4 = out_size / 4;

    rgcn_zero<<<(n4 + 255) / 256, 256, 0, stream>>>((float4*)out, n4);

    const int rowGroups = N / 128;             // 8 row tiles (of 16 rows) per block
    rgcn_basis_gemm<<<rowGroups * 4 * NBASE, 256, 0, stream>>>(text, bases, t);

    rgcn_edge_scatter<<<(E + 7) / 8, 256, 0, stream>>>(t, src, dst, rel, comp, out, E);

    rgcn_bias_relu<<<(n4 + 255) / 256, 256, 0, stream>>>(out, bias, n4);
}